// HilbertPatchedEmbedding_88871463289483
// MI455X (gfx1250) — compile-verified
//
#include <hip/hip_runtime.h>

// HilbertPatchedEmbedding for MI455X (gfx1250).
// Pure data-movement problem: ~525 MB HBM traffic, ~23 us at 23.3 TB/s peak.
// Strategy: async global->LDS staging of 32 Hilbert-consecutive patches
// (spatially a 4x8 patch block -> good L2 line reuse), LDS transpose,
// fully coalesced nontemporal float4 stores with the pos_emb add fused.

#define NBLK   32      // patch grid is NBLK x NBLK
#define PSZ    16      // patch size
#define CH     4       // channels
#define IMG    512     // NBLK*PSZ
#define EMB    1024    // C*P*P
#define SEQ    1025
#define NB     64      // batch
#define TJ     32      // patches per workgroup (one Hilbert tile)
#define STRIDE 1028    // EMB + 4 pad floats: 16B-aligned rows, <=2-way bank conflict

typedef float v4f __attribute__((ext_vector_type(4)));  // native vector: OK for
                                                        // __builtin_nontemporal_*

__device__ __forceinline__ int hilbert_g(int d) {
    // Standard Hilbert d2xy for n=32; returns gx*32 + gy.  Uniform -> SALU.
    int x = 0, y = 0, t = d;
    #pragma unroll
    for (int s = 1; s < NBLK; s *= 2) {
        int rx = 1 & (t >> 1);
        int ry = 1 & (t ^ rx);
        if (ry == 0) {
            if (rx == 1) { x = s - 1 - x; y = s - 1 - y; }
            int tmp = x; x = y; y = tmp;
        }
        x += s * rx;
        y += s * ry;
        t >>= 2;
    }
    return x * NBLK + y;
}

__global__ __launch_bounds__(256)
void hilbert_embed_main(const float* __restrict__ x,
                        const float* __restrict__ cls,
                        const float* __restrict__ pos,
                        float* __restrict__ out) {
    __shared__ float lds[TJ * STRIDE];   // 131,584 B -> 2 WGs / 320KB WGP

    const int tid  = threadIdx.x;
    const int b    = blockIdx.x >> 5;         // batch
    const int tile = blockIdx.x & 31;
    const int j0   = tile * TJ;               // first Hilbert index of tile

    // Real LDS byte offset of the staging buffer.  The ptrtoint both gives the
    // correct VDST value for the async op (generic-pointer low 32 bits == LDS
    // offset) and *captures* the array so the compiler knows the asm writes it.
    const unsigned lds_base = (unsigned)(uintptr_t)(void*)lds;

    // ---- Phase 1: async-copy 32 patches into LDS ------------------------
    // Iteration `it` moves patch j0+it: 256 lanes x 16B = one 4KB patch.
    const int q  = tid & 3;          // float4 within a 16-float patch row
    const int px = (tid >> 2) & 15;  // row within patch
    const int c  = tid >> 6;         // channel
    const unsigned lane_lds  = lds_base + (unsigned)((c * 256 + px * 16 + q * 4) * 4);
    const unsigned lane_goff =
        (unsigned)(((long long)b * (CH * IMG * IMG) +
                    c * (IMG * IMG) + px * IMG + q * 4) * 4);   // global bytes (<256MB)

    for (int it = 0; it < TJ; ++it) {
        const int g  = hilbert_g(j0 + it);       // uniform -> SALU
        const int gx = g >> 5, gy = g & 31;
        unsigned lds_off = lane_lds + (unsigned)(it * (STRIDE * 4));
        unsigned goff    = lane_goff + (unsigned)((gx * PSZ * IMG + gy * PSZ) * 4);
        // GVS mode: addr = SGPR64(x) + VGPR_i32(goff); LDS dest = VGPR(lds_off).
        asm volatile("global_load_async_to_lds_b128 %0, %1, %2"
                     :
                     : "v"(lds_off), "v"(goff), "s"(x)
                     : "memory");
    }

    // ---- cls row (s=0) for this batch, done by tile 0 while DMA runs ----
    if (tile == 0) {
        const int e = tid * 4;
        const v4f cv = *(const v4f*)(cls + e);
        const v4f pv = *(const v4f*)(pos + e);
        v4f v = cv + pv;
        __builtin_nontemporal_store(v, (v4f*)(out + (size_t)b * (SEQ * EMB) + e));
    }

    asm volatile("s_wait_asynccnt 0" ::: "memory");
    __syncthreads();

    // ---- Phase 2: transposed, coalesced write-out with pos_emb add ------
    // out[b, 1+i, j0+jj] = lds[jj][i] + pos[1+i, j0+jj]
    const int jq = (tid & 7) * 4;   // float4 group along patch axis (32 jj total)
    const int i0 = tid >> 3;        // i lane, step 32
    float* __restrict__ outb = out + (size_t)b * (SEQ * EMB);

    #pragma unroll 8
    for (int i = i0; i < EMB; i += 32) {
        const size_t row = (size_t)(1 + i) * EMB + (size_t)(j0 + jq);
        const v4f p = *(const v4f*)(pos + row);   // L2-resident across batches
        v4f v;
        v.x = lds[(jq + 0) * STRIDE + i] + p.x;
        v.y = lds[(jq + 1) * STRIDE + i] + p.y;
        v.z = lds[(jq + 2) * STRIDE + i] + p.z;
        v.w = lds[(jq + 3) * STRIDE + i] + p.w;
        // wave: 4 x 128B contiguous segments; NT keeps pos_emb/x hot in L2
        __builtin_nontemporal_store(v, (v4f*)(outb + row));
    }
}

extern "C" void kernel_launch(void* const* d_in, const int* in_sizes, int n_in,
                              void* d_out, int out_size, void* d_ws, size_t ws_size,
                              hipStream_t stream) {
    const float* x   = (const float*)d_in[0];   // (64, 4, 512, 512)
    const float* cls = (const float*)d_in[1];   // (1, 1, 1024)
    const float* pos = (const float*)d_in[2];   // (1, 1025, 1024)
    float* out = (float*)d_out;                 // (64, 1025, 1024)

    hipLaunchKernelGGL(hilbert_embed_main,
                       dim3(NB * (NBLK * NBLK / TJ)), dim3(256), 0, stream,
                       x, cls, pos, out);
}